// PcnDecoder_45406394253983
// MI455X (gfx1250) — compile-verified
//
#include <hip/hip_runtime.h>

typedef __attribute__((ext_vector_type(2))) float    v2f;
typedef __attribute__((ext_vector_type(8))) float    v8f;
typedef __attribute__((ext_vector_type(4))) uint32_t v4u;
typedef __attribute__((ext_vector_type(4))) int      v4i;
typedef __attribute__((ext_vector_type(8))) int      v8i;

#define MT        64                       // rows of h per block
#define NTHREADS  512                      // 16 waves
#define H1_PITCH  522                      // 512 + 10 dwords pad
#define KC        32                       // K-chunk per TDM transfer
#define BT_PITCH  (KC + 2)                 // TDM pad: 2 dwords per 32 -> pitch 34
#define H1_FLOATS (MT * H1_PITCH)
#define BUF_FLOATS (512 * BT_PITCH)        // one padded [n][k] panel
#define SMEM_FLOATS (2 * BUF_FLOATS + H1_FLOATS)   // bufs first (low LDS offsets)

// ---------------- Stage A: tiny linear chain + misc precompute ----------------

__global__ void pcn_gemm_naive(const float* __restrict__ A, const float* __restrict__ W,
                               const float* __restrict__ bias, float* __restrict__ out,
                               float* __restrict__ out2, int M, int K, int N) {
  int idx = blockIdx.x * 256 + threadIdx.x;
  if (idx >= M * N) return;
  int m = idx / N, n = idx - m * N;
  float s = bias[n];
  const float* a = A + (size_t)m * K;
  const float* w = W + n;
  for (int k = 0; k < K; ++k) s += a[k] * w[(size_t)k * N];
  out[idx] = s;
  if (out2) out2[idx] = s;
}

__global__ void pcn_precompute_misc(const float* __restrict__ x, const float* __restrict__ cw1,
                                    const float* __restrict__ cb1,
                                    float* __restrict__ glob, float* __restrict__ gridp) {
  int t = blockIdx.x * 256 + threadIdx.x;
  if (t < 4096) {                       // glob: 8 x 512 (x @ cw1[5:] + cb1)
    int b = t >> 9, n = t & 511;
    float s = cb1[n];
    const float* xb = x + (size_t)b * 1024;
    for (int k = 0; k < 1024; ++k) s += xb[k] * cw1[(size_t)(5 + k) * 512 + n];
    glob[t] = s;
  } else if (t < 12288) {               // gridp: 16 x 512
    int i2 = t - 4096;
    int g = i2 >> 9, n = i2 & 511;
    int gi = g >> 2, gj = g & 3;
    const float step = 0.1f / 3.0f;     // linspace(-0.05, 0.05, 4)
    float gx = -0.05f + gi * step;
    float gy = -0.05f + gj * step;
    gridp[i2] = gx * cw1[n] + gy * cw1[512 + n];
  }
}

// dst[n][k] = src[k][n] for 512x512 (LDS-tiled, coalesced both sides)
__global__ void pcn_transpose512(const float* __restrict__ src, float* __restrict__ dst) {
  __shared__ float tile[32][33];
  int bx = blockIdx.x, by = blockIdx.y;              // 16x16 grid of 32x32 tiles
  int tx = threadIdx.x & 31, ty = threadIdx.x >> 5;  // 256 threads = 32x8
  for (int i = 0; i < 32; i += 8)
    tile[ty + i][tx] = src[(size_t)(by * 32 + ty + i) * 512 + bx * 32 + tx];
  __syncthreads();
  for (int i = 0; i < 32; i += 8)
    dst[(size_t)(bx * 32 + ty + i) * 512 + by * 32 + tx] = tile[tx][ty + i];
}

// ---------------- TDM: DMA a [512 n][KC k] panel of cw2T into LDS (padded) ----------------
__device__ __forceinline__ void tdm_load_cw2T_chunk(const float* cw2T, int chunk,
                                                    uint32_t lds_byte_off) {
  uint64_t ga = (uint64_t)(uintptr_t)(cw2T + chunk);      // tile start: (n=0, k=chunk)
  v4u g0;
  g0[0] = 1u;                                             // count=1 (valid user D#)
  g0[1] = lds_byte_off;                                   // lds_addr
  g0[2] = (uint32_t)ga;                                   // global_addr[31:0]
  g0[3] = (uint32_t)((ga >> 32) & 0x1FFFFFFu) | 0x80000000u; // addr[56:32] | type=2
  v8i g1;
  // data_size=4B | pad_enable | pad_interval=4 (every 32 dwords) | pad_amount=1 (2 dwords)
  g1[0] = (2 << 16) | (1 << 20) | (4 << 22) | (1 << 25);  // 0x03120000
  g1[1] = (int)(512u << 16);                              // tensor_dim0 = 512
  g1[2] = (int)(512u << 16);                              // tensor_dim1 = 512
  g1[3] = (int)((unsigned)KC << 16);                      // tile_dim0 = KC (contig k)
  g1[4] = 512;                                            // tile_dim1 = 512 (n rows)
  g1[5] = 512;                                            // tensor_dim0_stride = 512
  g1[6] = 0;
  g1[7] = 0;
  v4i z4 = {0, 0, 0, 0};
#if defined(__clang_major__) && (__clang_major__ >= 23)
  v8i z8 = {0, 0, 0, 0, 0, 0, 0, 0};
  __builtin_amdgcn_tensor_load_to_lds(g0, g1, z4, z4, z8, 0);
#else
  __builtin_amdgcn_tensor_load_to_lds(g0, g1, z4, z4, 0);
#endif
}

// ---------------- Stage B: fused fine decoder (WMMA f32 + TDM double buffer) ----------------
// Block: MT=64 rows of h (one batch), N=512. 16 waves, each 16 rows x 128 cols = 8 C tiles.
__global__ __launch_bounds__(NTHREADS)
void pcn_fine_kernel(const float* __restrict__ coarse,   // (8,1024,3) ws copy
                     const float* __restrict__ glob,     // (8,512) incl cb1
                     const float* __restrict__ gridp,    // (16,512)
                     const float* __restrict__ cw1,      // (1029,512): rows 2..4 = w_pt
                     const float* __restrict__ cw2T,     // (512,512) n-major transpose
                     const float* __restrict__ cb2,      // (512,)
                     const float* __restrict__ cw3,      // (512,3)
                     const float* __restrict__ cb3,      // (3,)
                     float* __restrict__ fine) {         // (8,16384,3)
  extern __shared__ float smem[];
  float* bufs = smem;                      // [2][512][BT_PITCH] cw2T panels at offset 0
  float* h1   = smem + 2 * BUF_FLOATS;     // [MT][H1_PITCH] above the buffers

  const int tid  = threadIdx.x;
  const int lane = tid & 31;
  const int wid  = tid >> 5;
  const int b    = blockIdx.y;
  const int rowBase = blockIdx.x * MT;     // fine-row base within batch

  const uint32_t BUF_BYTES = (uint32_t)(BUF_FLOATS * sizeof(float));

  // Kick off DMA of cw2T chunk 0 while we build h1 (overlapped).
  if (wid == 0) tdm_load_cw2T_chunk(cw2T, 0, 0u);

  // ---- Phase 1: h1 = relu(grid + pt + glob(+cb1)) -> LDS ----
  for (int i = tid; i < MT * 512; i += NTHREADS) {
    int r = i >> 9, n = i & 511;
    int f = rowBase + r;
    int c = f >> 4, g = f & 15;
    const float* cc = coarse + ((size_t)b * 1024 + c) * 3;
    float pre = gridp[g * 512 + n] + glob[b * 512 + n]
              + cc[0] * cw1[2 * 512 + n]
              + cc[1] * cw1[3 * 512 + n]
              + cc[2] * cw1[4 * 512 + n];
    h1[r * H1_PITCH + n] = pre > 0.f ? pre : 0.f;
  }
  if (wid == 0) __builtin_amdgcn_s_wait_tensorcnt(0);
  __syncthreads();

  // ---- Phase 2: acc = h1 @ cw2 via V_WMMA_F32_16X16X4_F32, TDM double-buffered ----
  v8f acc[8] = {};
  const int wrow = (wid >> 2) * 16;        // 0,16,32,48
  const int wcol = (wid & 3) * 128;        // 0..384
  const int lmod = lane & 15;
  const int lhi2 = (lane >> 4) * 2;        // fragment K rotation per ISA layout

  // Hoisted lane-dependent bases (pure SSA -> stays in LDS address space).
  const float* h1row = h1 + (wrow + lmod) * H1_PITCH + lhi2;       // A frags
  const float* bt0   = bufs + (wcol + lmod) * BT_PITCH + lhi2;     // B frags buf0
  const float* bt1   = bt0 + BUF_FLOATS;                           // B frags buf1

  auto compute_chunk = [&](const float* __restrict__ bt, int ci) {
#pragma unroll 4
    for (int k0 = 0; k0 < KC; k0 += 4) {
      // A frag: 16x4 of h1 (M = lane&15, K pair) -> one b64 LDS load
      v2f a = *(const v2f*)&h1row[ci * KC + k0];
#pragma unroll
      for (int t = 0; t < 8; ++t) {
        // B frag: 4x16 (K pair contiguous in k-major panel) -> one b64 LDS load
        v2f bf = *(const v2f*)&bt[t * 16 * BT_PITCH + k0];
        acc[t] = __builtin_amdgcn_wmma_f32_16x16x4_f32(
            /*neg_a=*/false, a, /*neg_b=*/false, bf,
            /*c_mod=*/(short)0, acc[t], /*reuse_a=*/false, /*reuse_b=*/false);
      }
    }
  };

  for (int ci = 0; ci < 512 / KC; ci += 2) {
    // even chunk ci: compute from buf0, prefetch ci+1 into buf1
    if (wid == 0) tdm_load_cw2T_chunk(cw2T, (ci + 1) * KC, BUF_BYTES);
    compute_chunk(bt0, ci);
    if (wid == 0) __builtin_amdgcn_s_wait_tensorcnt(0);
    __syncthreads();

    // odd chunk ci+1: compute from buf1, prefetch ci+2 into buf0
    if (wid == 0 && ci + 2 < 512 / KC) tdm_load_cw2T_chunk(cw2T, (ci + 2) * KC, 0u);
    compute_chunk(bt1, ci + 1);
    if (wid == 0) __builtin_amdgcn_s_wait_tensorcnt(0);
    __syncthreads();
  }

  // ---- Phase 3a: h2 = relu(acc + cb2) -> reuse h1 buffer ----
#pragma unroll
  for (int t = 0; t < 8; ++t) {
    int n = wcol + t * 16 + lmod;
    float bias = cb2[n];
#pragma unroll
    for (int v = 0; v < 8; ++v) {
      int r = wrow + v + 8 * (lane >> 4);  // C layout: VGPR v -> M=v (+8 for hi lanes)
      float h = acc[t][v] + bias;
      h1[r * H1_PITCH + n] = h > 0.f ? h : 0.f;
    }
  }
  __syncthreads();

  // ---- Phase 3b: fine = h2 @ cw3 + cb3 + coarse skip ----
  float* part = bufs;                      // reuse: [MT][8][3]
  {
    int r = tid >> 3, seg = tid & 7;       // each thread: 64 of 512 K terms
    const float* hr = h1 + r * H1_PITCH + seg * 64;
    float p0 = 0.f, p1 = 0.f, p2 = 0.f;
#pragma unroll 4
    for (int n = 0; n < 64; ++n) {
      float h = hr[n];
      const float* w = cw3 + (size_t)(seg * 64 + n) * 3;
      p0 += h * w[0]; p1 += h * w[1]; p2 += h * w[2];
    }
    int base = (r * 8 + seg) * 3;
    part[base + 0] = p0; part[base + 1] = p1; part[base + 2] = p2;
  }
  __syncthreads();
  if (tid < MT * 3) {
    int r = tid / 3, j = tid - r * 3;
    float s = cb3[j];
#pragma unroll
    for (int sgm = 0; sgm < 8; ++sgm) s += part[(r * 8 + sgm) * 3 + j];
    int f = rowBase + r;
    int c = f >> 4;
    s += coarse[((size_t)b * 1024 + c) * 3 + j];
    fine[((size_t)b * 16384 + f) * 3 + j] = s;
  }
}

// ---------------- Launch ----------------
extern "C" void kernel_launch(void* const* d_in, const int* in_sizes, int n_in,
                              void* d_out, int out_size, void* d_ws, size_t ws_size,
                              hipStream_t stream) {
  const float* x   = (const float*)d_in[0];
  const float* w1  = (const float*)d_in[1];
  const float* b1  = (const float*)d_in[2];
  const float* w2  = (const float*)d_in[3];
  const float* b2  = (const float*)d_in[4];
  const float* w3  = (const float*)d_in[5];
  const float* b3  = (const float*)d_in[6];
  const float* cw1 = (const float*)d_in[7];
  const float* cb1 = (const float*)d_in[8];
  const float* cw2 = (const float*)d_in[9];
  const float* cb2 = (const float*)d_in[10];
  const float* cw3 = (const float*)d_in[11];
  const float* cb3 = (const float*)d_in[12];

  float* out        = (float*)d_out;
  float* coarse_out = out;                  // 8*1024*3 = 24576
  float* fine_out   = out + 24576;          // 8*16384*3 = 393216

  float* ws        = (float*)d_ws;
  float* t1        = ws;                    // 8*1024
  float* t2        = ws + 8192;             // 8*1024
  float* coarse_ws = ws + 16384;            // 24576
  float* glob      = ws + 40960;            // 8*512
  float* gridp     = ws + 45056;            // 16*512
  float* cw2T      = ws + 53248;            // 512*512 transposed cw2

  // coarse = ((x@w1+b1)@w2+b2)@w3+b3
  pcn_gemm_naive<<<(8 * 1024 + 255) / 256, 256, 0, stream>>>(x,  w1, b1, t1, nullptr, 8, 1024, 1024);
  pcn_gemm_naive<<<(8 * 1024 + 255) / 256, 256, 0, stream>>>(t1, w2, b2, t2, nullptr, 8, 1024, 1024);
  pcn_gemm_naive<<<(8 * 3072 + 255) / 256, 256, 0, stream>>>(t2, w3, b3, coarse_ws, coarse_out, 8, 1024, 3072);
  pcn_precompute_misc<<<48, 256, 0, stream>>>(x, cw1, cb1, glob, gridp);
  pcn_transpose512<<<dim3(16, 16), 256, 0, stream>>>(cw2, cw2T);

  size_t smem = SMEM_FLOATS * sizeof(float);   // ~267 KB dynamic LDS
  hipFuncSetAttribute((const void*)pcn_fine_kernel,
                      hipFuncAttributeMaxDynamicSharedMemorySize, (int)smem);
  dim3 grid(16384 / MT, 8);                 // 256 m-tiles x B
  pcn_fine_kernel<<<grid, NTHREADS, smem, stream>>>(coarse_ws, glob, gridp, cw1, cw2T,
                                                    cb2, cw3, cb3, fine_out);
}